// GATEncoder_57887569215652
// MI455X (gfx1250) — compile-verified
//
#include <hip/hip_runtime.h>

// ---------------------------------------------------------------------------
// GAT encoder for MI455X (gfx1250, wave32).
// Key insight: reference einsum 'nhll,nhld->nhld' uses only the DIAGONAL
// attention coefficient, so we never materialize the (N,H,A,A) tensor —
// only per-row online-softmax (max,sum) and the diagonal score.
// GEMM runs on the matrix pipe via V_WMMA_F32_16X16X4_F32 (exact fp32).
// ---------------------------------------------------------------------------

typedef float v2f __attribute__((ext_vector_type(2)));
typedef float v8f __attribute__((ext_vector_type(8)));

namespace gat {
constexpr int Nn    = 2;
constexpr int Ares  = 96;      // residues (L)
constexpr int ATOMS = 14;
constexpr int A     = Ares * ATOMS;  // 1344 atoms / protein
constexpr int D     = 128;
constexpr int H     = 8;
constexpr int E     = H * D;   // 1024
constexpr int NA    = Nn * A;  // 2688 rows
constexpr float LN_EPS = 1e-5f;
// jnp.finfo(f32).min * 0.5
constexpr float NEGBIG = -1.7014118346046923e38f;

__device__ __forceinline__ float lrelu02(float x) {
  // leaky_relu(x, 0.2) == max(x, 0.2*x) for all x
  return fmaxf(x, 0.2f * x);
}
} // namespace gat

// ---------------------------------------------------------------------------
// Kernel 1: fp = X(2688x128) * W^T(128x1024) via V_WMMA_F32_16X16X4_F32.
// One wave per 16x16 output tile; K=128 -> 32 WMMA steps.
// A-frag 16x4 f32 layout: lane<16 holds {K=k0,k0+1}, lane>=16 {k0+2,k0+3},
// row = lane&15.  B (4x16) mirrors it with col = lane&15 (B[k][n] = W[n][k],
// so each lane loads a contiguous float2 of W).  C/D: vgpr r, lane l ->
// out[r + 8*(l>>4)][l&15].
// ---------------------------------------------------------------------------
__global__ void __launch_bounds__(256)
gat_gemm_wmma(const float* __restrict__ X, const float* __restrict__ W,
              float* __restrict__ FP) {
  using namespace gat;
  const int wid   = blockIdx.x * 8 + (threadIdx.x >> 5);  // 0..10751
  const int lane  = threadIdx.x & 31;
  const int tileM = wid % (NA / 16);   // 0..167
  const int tileN = wid / (NA / 16);   // 0..63
  const int half  = lane >> 4;
  const int l15   = lane & 15;

  const float* arow = X + (tileM * 16 + l15) * D;  // A row for this lane
  const float* brow = W + (tileN * 16 + l15) * D;  // W row (= B column)

  v8f acc = {};
#pragma unroll
  for (int k0 = 0; k0 < D; k0 += 4) {
    v2f a = *(const v2f*)(arow + k0 + 2 * half);
    v2f b = *(const v2f*)(brow + k0 + 2 * half);
    acc = __builtin_amdgcn_wmma_f32_16x16x4_f32(
        /*neg_a=*/false, a, /*neg_b=*/false, b,
        /*c_mod=*/(short)0, acc, /*reuse_a=*/false, /*reuse_b=*/false);
  }

  const int col = tileN * 16 + l15;
#pragma unroll
  for (int r = 0; r < 8; ++r) {
    const int row = tileM * 16 + r + 8 * half;
    FP[row * E + col] = acc[r];
  }
}

// ---------------------------------------------------------------------------
// Kernel 2: masked per-(n,a,h) dots with s_src/s_tag.  One wave per (n,a,h),
// 4 floats per lane, wave reduction by shuffles.  Results stored transposed
// (na, h) so kernel 3 reads 8 contiguous floats per atom.
// ---------------------------------------------------------------------------
__global__ void __launch_bounds__(256)
gat_score(const float* __restrict__ FP, const int* __restrict__ mask,
          const float* __restrict__ s_src, const float* __restrict__ s_tag,
          float* __restrict__ srcT, float* __restrict__ tagT) {
  using namespace gat;
  const int w    = blockIdx.x * 8 + (threadIdx.x >> 5);  // 0..21503
  const int lane = threadIdx.x & 31;
  const int h    = w & (H - 1);
  const int na   = w >> 3;
  const float mk = mask[na] ? 1.0f : 0.0f;

  const int d0 = lane * 4;
  const float* f = FP + na * E + h * D + d0;
  const float* ps = s_src + h * D + d0;
  const float* pt = s_tag + h * D + d0;

  float as = 0.f, at = 0.f;
#pragma unroll
  for (int i = 0; i < 4; ++i) {
    const float v = f[i] * mk;
    as += v * ps[i];
    at += v * pt[i];
  }
#pragma unroll
  for (int off = 16; off > 0; off >>= 1) {
    as += __shfl_xor(as, off, 32);
    at += __shfl_xor(at, off, 32);
  }
  if (lane == 0) {
    srcT[na * H + h] = as;
    tagT[na * H + h] = at;
  }
}

// ---------------------------------------------------------------------------
// Kernel 3: diagonal attention via online softmax.  One wave per (n,i) row,
// all 8 heads share a single edge_mask row read (8x mask-bandwidth saving).
// Sentinel NEGBIG reproduces JAX softmax exactly (all-masked row -> uniform).
// ---------------------------------------------------------------------------
__global__ void __launch_bounds__(256)
gat_attdiag(const float* __restrict__ srcT, const float* __restrict__ tagT,
            const int* __restrict__ edge_mask, float* __restrict__ attT) {
  using namespace gat;
  const int w    = blockIdx.x * 8 + (threadIdx.x >> 5);  // 0..2687
  const int lane = threadIdx.x & 31;
  const int n = w / A;
  const int i = w % A;

  float sr[H];
#pragma unroll
  for (int h = 0; h < H; ++h) sr[h] = srcT[(n * A + i) * H + h];

  float m[H], su[H], dg[H];
#pragma unroll
  for (int h = 0; h < H; ++h) { m[h] = NEGBIG; su[h] = 0.f; dg[h] = NEGBIG; }

  const int* em = edge_mask + (n * A + i) * A;
  for (int j = lane; j < A; j += 32) {
    const float e = em[j] ? 1.0f : 0.0f;
    const float* tg = tagT + (n * A + j) * H;
#pragma unroll
    for (int h = 0; h < H; ++h) {
      float s = lrelu02(sr[h] + tg[h]);
      s = e * s + (1.0f - e) * NEGBIG;     // masked -> NEGBIG
      if (j == i) dg[h] = s;
      const float mn = fmaxf(m[h], s);
      su[h] = su[h] * __expf(m[h] - mn) + __expf(s - mn);
      m[h] = mn;
    }
  }
  // cross-lane merge of (m,sum), max-merge of diag
#pragma unroll
  for (int off = 16; off > 0; off >>= 1) {
#pragma unroll
    for (int h = 0; h < H; ++h) {
      const float om = __shfl_xor(m[h], off, 32);
      const float os = __shfl_xor(su[h], off, 32);
      const float mn = fmaxf(m[h], om);
      su[h] = su[h] * __expf(m[h] - mn) + os * __expf(om - mn);
      m[h] = mn;
      dg[h] = fmaxf(dg[h], __shfl_xor(dg[h], off, 32));
    }
  }
  if (lane == 0) {
#pragma unroll
    for (int h = 0; h < H; ++h)
      attT[(n * A + i) * H + h] = __expf(dg[h] - m[h]) / su[h];
  }
}

// ---------------------------------------------------------------------------
// Kernel 4: fo = att*fp (masked), LayerNorm per head, head-mean via LDS
// ds_add_f32 atomics, skip + bias.  One block (8 waves = 8 heads) per atom.
// ---------------------------------------------------------------------------
__global__ void __launch_bounds__(256)
gat_output(const float* __restrict__ FP, const float* __restrict__ attT,
           const int* __restrict__ mask, const float* __restrict__ xin,
           const float* __restrict__ g, const float* __restrict__ b,
           const float* __restrict__ bias, float* __restrict__ xout) {
  using namespace gat;
  __shared__ float accd[D];
  const int na   = blockIdx.x;
  const int tid  = threadIdx.x;
  const int h    = tid >> 5;
  const int lane = tid & 31;
  const int d0   = lane * 4;

  if (tid < D) accd[tid] = 0.0f;
  __syncthreads();

  const float mk  = mask[na] ? 1.0f : 0.0f;
  const float att = attT[na * H + h];
  const float* f  = FP + na * E + h * D + d0;

  float v[4];
  float s = 0.f, ss = 0.f;
#pragma unroll
  for (int i = 0; i < 4; ++i) {
    v[i] = f[i] * mk * att;
    s += v[i];
    ss += v[i] * v[i];
  }
#pragma unroll
  for (int off = 16; off > 0; off >>= 1) {
    s  += __shfl_xor(s, off, 32);
    ss += __shfl_xor(ss, off, 32);
  }
  const float mu  = s * (1.0f / D);
  const float var = fmaxf(ss * (1.0f / D) - mu * mu, 0.0f);
  const float rstd = rsqrtf(var + LN_EPS);

#pragma unroll
  for (int i = 0; i < 4; ++i) {
    const float nv = (v[i] - mu) * rstd * g[d0 + i] + b[d0 + i];
    atomicAdd(&accd[d0 + i], nv);   // ds_add_f32
  }
  __syncthreads();

  if (tid < D) {
    xout[na * D + tid] = accd[tid] * (1.0f / H) + xin[na * D + tid] + bias[tid];
  }
}

// ---------------------------------------------------------------------------
// Kernel 5: masked residue-sum readout: (N, L, D) = sum over 14 atoms.
// ---------------------------------------------------------------------------
__global__ void __launch_bounds__(256)
gat_readout(const float* __restrict__ x, const int* __restrict__ mask,
            float* __restrict__ out) {
  using namespace gat;
  const int idx = blockIdx.x * 256 + threadIdx.x;   // N*L*D = 24576
  if (idx >= Nn * Ares * D) return;
  const int d  = idx & (D - 1);
  const int nl = idx >> 7;          // n*L + l
  const int n  = nl / Ares;
  const int l  = nl % Ares;
  float acc = 0.f;
#pragma unroll
  for (int t = 0; t < ATOMS; ++t) {
    const int a = l * ATOMS + t;
    const float mk = mask[n * A + a] ? 1.0f : 0.0f;
    acc += mk * x[(n * A + a) * D + d];
  }
  out[idx] = acc;
}

// ---------------------------------------------------------------------------
// Host side: 3 layers x {gemm, score, attdiag, output} + readout.
// Inputs: 0 feats(f32) 1 mask(i) 2 edge_mask(i) 3 W(f32) 4 s_src 5 s_tag
//         6 bias 7 ln_g 8 ln_b
// ---------------------------------------------------------------------------
extern "C" void kernel_launch(void* const* d_in, const int* in_sizes, int n_in,
                              void* d_out, int out_size, void* d_ws, size_t ws_size,
                              hipStream_t stream) {
  using namespace gat;
  const float* feats = (const float*)d_in[0];
  const int*   mask  = (const int*)d_in[1];
  const int*   emask = (const int*)d_in[2];
  const float* W     = (const float*)d_in[3];
  const float* ssrc  = (const float*)d_in[4];
  const float* stag  = (const float*)d_in[5];
  const float* bias  = (const float*)d_in[6];
  const float* ln_g  = (const float*)d_in[7];
  const float* ln_b  = (const float*)d_in[8];

  float* ws   = (float*)d_ws;
  float* FP   = ws;                 // NA*E      = 2,752,512 f
  float* xA   = FP + NA * E;        // NA*D
  float* xB   = xA + NA * D;        // NA*D
  float* srcT = xB + NA * D;        // NA*H
  float* tagT = srcT + NA * H;      // NA*H
  float* attT = tagT + NA * H;      // NA*H

  const float* xc = feats;
  for (int l = 0; l < 3; ++l) {
    float* xn = (l & 1) ? xB : xA;
    gat_gemm_wmma<<<(NA / 16) * (E / 16) / 8, 256, 0, stream>>>(
        xc, W + (size_t)l * E * D, FP);
    gat_score<<<NA * H / 8, 256, 0, stream>>>(
        FP, mask, ssrc + l * H * D, stag + l * H * D, srcT, tagT);
    gat_attdiag<<<NA / 8, 256, 0, stream>>>(srcT, tagT, emask, attT);
    gat_output<<<NA, 256, 0, stream>>>(
        FP, attT, mask, xc, ln_g + l * D, ln_b + l * D, bias + l * D, xn);
    xc = xn;
  }
  gat_readout<<<(Nn * Ares * D + 255) / 256, 256, 0, stream>>>(xc, mask,
                                                               (float*)d_out);
}